// WENONetwork_1881195676124
// MI455X (gfx1250) — compile-verified
//
#include <hip/hip_runtime.h>
#include <hip/hip_bf16.h>
#include <math.h>

typedef __attribute__((ext_vector_type(2))) float v2f;
typedef __attribute__((ext_vector_type(8))) float v8f;

#define TILE   256          // output points per block
#define BLOCK  256          // 8 wave32
#define HW     (TILE + 6)   // h positions p in [j0, j0+TILE+5]
#define HSTR   264          // padded LDS stride for h arrays
#define UUW    272          // sUU holds uu[j0-2 .. j0+TILE+8] (267 used)

__global__ __launch_bounds__(BLOCK) void weno5_wmma_kernel(
    const float* __restrict__ u,   // (N,2) row-major
    const float* __restrict__ w1,  // (6,1,5)
    const float* __restrict__ b1,  // (6,)
    const float* __restrict__ w2,  // (3,6,5)
    const float* __restrict__ b2,  // (3,)
    const float* __restrict__ ep,  // scalar e
    const int*   __restrict__ lp,  // scalar l
    float* __restrict__ out, int N, int NOUT)
{
    __shared__ float sUU[UUW];
    __shared__ float sHl[6][HSTR];
    __shared__ float sHr[6][HSTR];
    __shared__ float sCr[3][TILE];
    __shared__ float sCl[3][TILE];
    __shared__ float sW1[30], sB1[6], sW2[90], sB2[3];

    const int tid = threadIdx.x;
    const int j0  = blockIdx.x * TILE;
    const int col = lp[0] - 1;
    const float e = ep[0];

    // ---- phase 1: weights + uu tile into LDS (zero outside [0,N)) ----
    if (tid < 30) sW1[tid] = w1[tid];
    if (tid < 6)  sB1[tid] = b1[tid];
    if (tid < 90) sW2[tid] = w2[tid];
    if (tid < 3)  sB2[tid] = b2[tid];
    for (int i = tid; i < 267; i += BLOCK) {
        int g = j0 - 2 + i;                       // uu global index
        sUU[i] = (g >= 0 && g < N) ? u[2 * g + col] : 0.0f;
    }
    __syncthreads();

    // ---- phase 2: h_l / h_r = relu(conv1) with exact SAME-padding per view ----
    // x_l[q] = uu[q]   valid for 0 <= q <= N-2  (uu[:-1])
    // x_r[q] = uu[q+1] valid for 0 <= q <= N-2  (uu[1:])
    for (int idx = tid; idx < HW; idx += BLOCK) {
        int p = j0 + idx;
        float xl[5], xr[5];
        #pragma unroll
        for (int k = 0; k < 5; ++k) {
            int q = p + k - 2;                    // x-index
            float vl = sUU[idx + k];              // uu[q]   (0 if q<0 or q>=N)
            float vr = sUU[idx + k + 1];          // uu[q+1] (0 if q+1>=N)
            xl[k] = (q <= N - 2) ? vl : 0.0f;     // exclude uu[N-1] from x_l
            xr[k] = (q >= 0)     ? vr : 0.0f;     // exclude uu[0]   from x_r
        }
        #pragma unroll
        for (int c = 0; c < 6; ++c) {
            float al = sB1[c], ar = sB1[c];
            #pragma unroll
            for (int k = 0; k < 5; ++k) {
                al = fmaf(sW1[c * 5 + k], xl[k], al);
                ar = fmaf(sW1[c * 5 + k], xr[k], ar);
            }
            sHl[c][idx] = fmaxf(al, 0.0f);
            sHr[c][idx] = fmaxf(ar, 0.0f);
        }
    }
    __syncthreads();

    // ---- phase 3: conv2 as f32 WMMA GEMM  D(16pts x 16ch) += A(16x4) * B(4x16) ----
    // A layout (ISA 7.12.2, 32-bit A 16x4): lane M = l&15, VGPR r holds K = r + 2*(l>>4)
    // B layout (symmetric striping):        lane N = l&15, VGPR r holds K = r + 2*(l>>4)
    // C/D layout: VGPR r at lane l -> (M = r + 8*(l>>4), N = l&15)
    const int lane   = tid & 31;
    const int waveId = tid >> 5;          // 0..7, each wave owns 32 output points
    const int m      = lane & 15;         // M row (output point in group)
    const bool hi    = (lane >> 4) != 0;
    const int n      = m;                 // N col (conv2 output channel, 0..2 used)
    const int jb0 = waveId * 32;
    const int jb1 = jb0 + 16;

    const float* hrBase = &sHr[0][0];
    const float* hlBase = &sHl[0][0];
    const int baseR0 = jb0 + m + 1;       // h_r index base (right uses j+1+t)
    const int baseL0 = jb0 + m;           // h_l index base (left uses j+t)
    const int baseR1 = jb1 + m + 1;
    const int baseL1 = jb1 + m;
    const bool nOK = (n < 3);
    const int  nw  = nOK ? n : 0;         // clamped w2 row (load always in-range)

    v8f accR0 = {}, accL0 = {}, accR1 = {}, accL1 = {};
    #pragma unroll
    for (int kc = 0; kc < 8; ++kc) {      // K = 30 padded to 32
        v2f aR0, aL0, aR1, aL1, bb;
        #pragma unroll
        for (int r = 0; r < 2; ++r) {
            // lane-half selects between two compile-time K values: ka (half 0), kb (half 1)
            const int  ka = 4 * kc + r;           // always < 30
            const int  kb = ka + 2;               // >= 30 only when kc == 7
            const bool vb = (kb < 30);
            const int  offA = (ka / 5) * HSTR + (ka % 5);                  // constant
            const int  offB = vb ? ((kb / 5) * HSTR + (kb % 5)) : 0;       // constant
            const int  kwB  = vb ? kb : 0;                                 // constant

            int  off  = hi ? offB : offA;         // one v_cndmask between constants
            int  kw   = hi ? kwB  : ka;
            bool vald = hi ? vb   : true;

            float bw = sW2[nw * 30 + kw];         // unconditional LDS load
            bb[r] = (vald && nOK) ? bw : 0.0f;    // select on value, no EXEC churn

            float hr0 = hrBase[off + baseR0];
            float hl0 = hlBase[off + baseL0];
            float hr1 = hrBase[off + baseR1];
            float hl1 = hlBase[off + baseL1];
            aR0[r] = vald ? hr0 : 0.0f;
            aL0[r] = vald ? hl0 : 0.0f;
            aR1[r] = vald ? hr1 : 0.0f;
            aL1[r] = vald ? hl1 : 0.0f;
        }
        accR0 = __builtin_amdgcn_wmma_f32_16x16x4_f32(false, aR0, false, bb, (short)0, accR0, false, false);
        accL0 = __builtin_amdgcn_wmma_f32_16x16x4_f32(false, aL0, false, bb, (short)0, accL0, false, false);
        accR1 = __builtin_amdgcn_wmma_f32_16x16x4_f32(false, aR1, false, bb, (short)0, accR1, false, false);
        accL1 = __builtin_amdgcn_wmma_f32_16x16x4_f32(false, aL1, false, bb, (short)0, accL1, false, false);
    }
    if (nOK) {                            // divergence only after all WMMAs
        float bo = sB2[n];
        const int hb = hi ? 8 : 0;
        #pragma unroll
        for (int r = 0; r < 8; ++r) {
            int mr = r + hb;
            sCr[n][jb0 + mr] = 1.0f / (1.0f + __expf(-(accR0[r] + bo)));
            sCl[n][jb0 + mr] = 1.0f / (1.0f + __expf(-(accL0[r] + bo)));
            sCr[n][jb1 + mr] = 1.0f / (1.0f + __expf(-(accR1[r] + bo)));
            sCl[n][jb1 + mr] = 1.0f / (1.0f + __expf(-(accL1[r] + bo)));
        }
    }
    __syncthreads();

    // ---- phase 4: fluxes, betas, omegas, final flux ----
    {
        const int t = tid;
        const int j = j0 + t;
        // ur_i = uu_right[j+i] = uu[j+i+1] = sUU[t+i+3]; ul_i = uu[j+i] = sUU[t+i+2]
        float ur1 = sUU[t + 4], ur2 = sUU[t + 5], ur3 = sUU[t + 6], ur4 = sUU[t + 7], ur5 = sUU[t + 8];
        float ul1 = sUU[t + 3], ul2 = sUU[t + 4], ul3 = sUU[t + 5], ul4 = sUU[t + 6], ul5 = sUU[t + 7];
        const float c16 = 1.0f / 6.0f, c1312 = 13.0f / 12.0f;
        auto sq = [](float x) { return x * x; };

        float fp0 = (11.f * ur3 - 7.f * ur4 + 2.f * ur5) * c16;
        float fp1 = (2.f * ur2 + 5.f * ur3 - ur4) * c16;
        float fp2 = (-ur1 + 5.f * ur2 + 2.f * ur3) * c16;
        float fn0 = (11.f * ul3 - 7.f * ul4 + 2.f * ul5) * c16;
        float fn1 = (2.f * ul2 + 5.f * ul3 - ul4) * c16;
        float fn2 = (-ul1 + 5.f * ul2 + 2.f * ul3) * c16;

        float bp0 = c1312 * sq(ur3 - 2.f * ur4 + ur5) + 0.25f * sq(3.f * ur3 - 4.f * ur4 + ur5);
        float bp1 = c1312 * sq(ur2 - 2.f * ur3 + ur4) + 0.25f * sq(ur2 - ur4);
        float bp2 = c1312 * sq(ur1 - 2.f * ur2 + ur3) + 0.25f * sq(ur1 - 4.f * ur2 + 3.f * ur3);
        float bn0 = c1312 * sq(ul3 - 2.f * ul4 + ul5) + 0.25f * sq(3.f * ul3 - 4.f * ul4 + ul5);
        float bn1 = c1312 * sq(ul2 - 2.f * ul3 + ul4) + 0.25f * sq(ul2 - ul4);
        float bn2 = c1312 * sq(ul1 - 2.f * ul2 + ul3) + 0.25f * sq(ul1 - 4.f * ul2 + 3.f * ul3);

        bp0 *= sCr[0][t]; bp1 *= sCr[1][t]; bp2 *= sCr[2][t];
        bn0 *= sCl[0][t]; bn1 *= sCl[1][t]; bn2 *= sCl[2][t];

        float fluxp, fluxn;
        {
            float rng2 = sq(bp2 - bp0);
            float e0 = sq(e + bp0), e1 = sq(e + bp1), e2 = sq(e + bp2);
            float w0 = 0.1f / e0 * (rng2 + e0);
            float w1v = 0.6f / e1 * (rng2 + e1);
            float w2v = 0.3f / e2 * (rng2 + e2);
            fluxp = (w0 * fp0 + w1v * fp1 + w2v * fp2) / (w0 + w1v + w2v);
        }
        {
            float rng2 = sq(bn2 - bn0);
            float e0 = sq(e + bn0), e1 = sq(e + bn1), e2 = sq(e + bn2);
            float w0 = 0.1f / e0 * (rng2 + e0);
            float w1v = 0.6f / e1 * (rng2 + e1);
            float w2v = 0.3f / e2 * (rng2 + e2);
            fluxn = (w0 * fn0 + w1v * fn1 + w2v * fn2) / (w0 + w1v + w2v);
        }
        if (j < NOUT) out[j] = fluxp - fluxn;
    }
}

extern "C" void kernel_launch(void* const* d_in, const int* in_sizes, int n_in,
                              void* d_out, int out_size, void* d_ws, size_t ws_size,
                              hipStream_t stream) {
    const float* u  = (const float*)d_in[0];
    const float* w1 = (const float*)d_in[1];
    const float* b1 = (const float*)d_in[2];
    const float* w2 = (const float*)d_in[3];
    const float* b2 = (const float*)d_in[4];
    const float* e  = (const float*)d_in[5];
    const int*   l  = (const int*)d_in[6];
    float* out = (float*)d_out;

    int N    = in_sizes[0] / 2;   // u is (N,2)
    int NOUT = out_size;          // N - 6
    int grid = (NOUT + TILE - 1) / TILE;
    weno5_wmma_kernel<<<grid, BLOCK, 0, stream>>>(u, w1, b1, w2, b2, e, l, out, N, NOUT);
}